// SAB_42339787604875
// MI455X (gfx1250) — compile-verified
//
#include <hip/hip_runtime.h>

typedef float v2f __attribute__((ext_vector_type(2)));
typedef float v8f __attribute__((ext_vector_type(8)));

#define BATCH 32
#define CHN   256
#define HDIM  56
#define WDIM  56
#define HWSZ  (HDIM * WDIM)          // 3136
#define HW4   (HWSZ / 4)             // 784

// ---------------------------------------------------------------------------
// Pass 1: channel-wise mean (via V_WMMA_F32_16X16X4_F32) + channel-wise max.
// One wave handles 16 spatial positions; 196 waves per batch image ->
// grid = BATCH*49 blocks of 128 threads (4 waves). No tail => EXEC all-ones.
//
// Per WMMA step (K=4): lane l (l<16) supplies channels {k,k+1} at position
// p0+l, lane l+16 supplies channels {k+2,k+3} at position p0+l. A = 1/256
// everywhere, so D[m,n] = mean over channels for position p0+n (exact fp32:
// 1/256 scaling is a power of two). Max is carried per-lane on the same
// loaded values, merged across half-waves with one shfl_xor(16).
// ---------------------------------------------------------------------------
__global__ __launch_bounds__(128) void sab_pool_kernel(
    const float* __restrict__ x,
    float* __restrict__ avg_out,
    float* __restrict__ max_out) {
  const int b    = blockIdx.x / 49;
  const int tile = blockIdx.x % 49;
  const int lane = threadIdx.x & 31;
  const int wave = threadIdx.x >> 5;

  const int p0  = (tile * 4 + wave) * 16;
  const int pos = p0 + (lane & 15);
  const int koff = (lane >> 4) * 2;   // lanes 0-15: ch k,k+1 ; lanes 16-31: ch k+2,k+3

  const float* base = x + (size_t)b * CHN * HWSZ + (size_t)koff * HWSZ + pos;

  v2f a;
  a[0] = 1.0f / 256.0f;
  a[1] = 1.0f / 256.0f;

  v8f acc0 = {};
  v8f acc1 = {};
  float mx = -__builtin_inff();

#pragma unroll
  for (int k = 0; k < CHN; k += 8) {
    // chain 0: channels k .. k+3
    float v0 = base[(size_t)k * HWSZ];
    float v1 = base[(size_t)k * HWSZ + HWSZ];
    // chain 1: channels k+4 .. k+7
    float v2 = base[(size_t)(k + 4) * HWSZ];
    float v3 = base[(size_t)(k + 4) * HWSZ + HWSZ];

    mx = fmaxf(mx, fmaxf(fmaxf(v0, v1), fmaxf(v2, v3)));

    v2f b0; b0[0] = v0; b0[1] = v1;
    v2f b1; b1[0] = v2; b1[1] = v3;

    acc0 = __builtin_amdgcn_wmma_f32_16x16x4_f32(
        false, a, false, b0, (short)0, acc0, false, false);
    acc1 = __builtin_amdgcn_wmma_f32_16x16x4_f32(
        false, a, false, b1, (short)0, acc1, false, false);
  }

  // merge the two half-wave channel partitions of the max
  mx = fmaxf(mx, __shfl_xor(mx, 16, 32));

  if (lane < 16) {
    const int q = b * HWSZ + pos;
    avg_out[q] = acc0[0] + acc1[0];   // row 0 of D: mean over all 256 channels
    max_out[q] = mx;
  }
}

// ---------------------------------------------------------------------------
// Pass 2: 3x3 conv (2 in-ch: avg, max), bias, sigmoid -> mask [B,HW].
// Tiny (100K pixels); avg/max live in L2. Weight layout OIHW: cw[0..8] for
// the avg channel, cw[9..17] for the max channel.
// ---------------------------------------------------------------------------
__global__ __launch_bounds__(256) void sab_mask_kernel(
    const float* __restrict__ avg,
    const float* __restrict__ mxp,
    const float* __restrict__ cw,
    const float* __restrict__ cb,
    float* __restrict__ mask) {
  const int idx = blockIdx.x * blockDim.x + threadIdx.x;
  if (idx >= BATCH * HWSZ) return;

  const int b = idx / HWSZ;
  const int p = idx - b * HWSZ;
  const int h = p / WDIM;
  const int w = p - h * WDIM;

  float acc = cb[0];
#pragma unroll
  for (int dh = -1; dh <= 1; ++dh) {
    const int hh = h + dh;
    if (hh < 0 || hh >= HDIM) continue;
#pragma unroll
    for (int dw = -1; dw <= 1; ++dw) {
      const int ww = w + dw;
      if (ww < 0 || ww >= WDIM) continue;
      const int q  = b * HWSZ + hh * WDIM + ww;
      const int wi = (dh + 1) * 3 + (dw + 1);
      acc += avg[q] * cw[wi] + mxp[q] * cw[9 + wi];
    }
  }
  mask[idx] = 1.0f / (1.0f + __expf(-acc));
}

// ---------------------------------------------------------------------------
// Pass 3: out[b,c,p] = x[b,c,p] * mask[b,p]. float4 streams (b128 vmem);
// mask (0.4 MB) is L2-resident, so this runs at the 2x102.8 MB stream rate.
// ---------------------------------------------------------------------------
__global__ __launch_bounds__(256) void sab_apply_kernel(
    const float* __restrict__ x,
    const float* __restrict__ mask,
    float* __restrict__ out) {
  const int total4 = BATCH * CHN * HW4;   // 6,422,528
  const int idx4 = blockIdx.x * blockDim.x + threadIdx.x;
  if (idx4 >= total4) return;

  const int p4 = idx4 % HW4;
  const int b  = idx4 / (CHN * HW4);

  const float4 xv = ((const float4*)x)[idx4];
  const float4 mv = ((const float4*)mask)[(size_t)b * HW4 + p4];

  float4 ov;
  ov.x = xv.x * mv.x;
  ov.y = xv.y * mv.y;
  ov.z = xv.z * mv.z;
  ov.w = xv.w * mv.w;
  ((float4*)out)[idx4] = ov;
}

extern "C" void kernel_launch(void* const* d_in, const int* in_sizes, int n_in,
                              void* d_out, int out_size, void* d_ws, size_t ws_size,
                              hipStream_t stream) {
  const float* x  = (const float*)d_in[0];   // [32,256,56,56]
  const float* cw = (const float*)d_in[1];   // [1,2,3,3] = 18 floats
  const float* cb = (const float*)d_in[2];   // [1]
  float* out = (float*)d_out;

  float* ws   = (float*)d_ws;
  float* avg  = ws;                          // BATCH*HWSZ floats
  float* mxp  = ws + BATCH * HWSZ;           // BATCH*HWSZ floats
  float* mask = ws + 2 * BATCH * HWSZ;       // BATCH*HWSZ floats

  // Pass 1: pooled avg/max via f32 WMMA + VALU max
  sab_pool_kernel<<<BATCH * 49, 128, 0, stream>>>(x, avg, mxp);

  // Pass 2: 3x3 conv + sigmoid -> mask
  const int npix = BATCH * HWSZ;
  sab_mask_kernel<<<(npix + 255) / 256, 256, 0, stream>>>(avg, mxp, cw, cb, mask);

  // Pass 3: gate x by mask (float4 streams)
  const int total4 = BATCH * CHN * HW4;
  sab_apply_kernel<<<(total4 + 255) / 256, 256, 0, stream>>>(x, mask, out);
}